// San_26860725469298
// MI455X (gfx1250) — compile-verified
//
#include <hip/hip_runtime.h>
#include <math.h>

typedef __attribute__((ext_vector_type(2))) float v2f;
typedef __attribute__((ext_vector_type(8))) float v8f;

#define NH 2048   // n_hidden (rows of M_h, M_p, W2)
#define HD 2048   // H
#define PD 2048   // P
#define G3 6144   // 3*H
#define NCHUNK 64 // row chunks for two-stage column reductions

// workspace layout (floats)
#define WS_SK        0
#define WS_COLSUM_H  (WS_SK + HD)
#define WS_MP_SUM    (WS_COLSUM_H + HD)
#define WS_W2T_MP    (WS_MP_SUM + PD)
#define WS_WMA       (WS_W2T_MP + HD)      // W_ih @ Mp_sum, K-half 0 (6144)
#define WS_WMB       (WS_WMA + G3)         // W_ih @ Mp_sum, K-half 1
#define WS_GHA       (WS_WMB + G3)         // W_hh @ sk, K-half 0
#define WS_GHB       (WS_GHA + G3)         // W_hh @ sk, K-half 1
#define WS_ACCUM     (WS_GHB + G3)         // 3
#define WS_SCAL      (WS_ACCUM + 3)        // [0]=alpha [1]=denom_p [2]=beta
#define WS_PART      (WS_SCAL + 8)         // NCHUNK * 2048 partials (reused)

// Stage 1: part[chunk][j] = sum_{i in chunk} w(i) * M[i][j]   (w==1 if wvec==nullptr)
__global__ void wcolsum_part_kernel(const float* __restrict__ M, const float* __restrict__ wvec,
                                    int wstride, float* __restrict__ part,
                                    int ncols, int rowsPerChunk) {
    const int j = blockIdx.x * blockDim.x + threadIdx.x;
    const int chunk = blockIdx.y;
    const int rbeg = chunk * rowsPerChunk;
    float s = 0.0f;
    if (wvec) {
        for (int i = rbeg; i < rbeg + rowsPerChunk; ++i)
            s += wvec[(size_t)i * wstride] * M[(size_t)i * ncols + j];
    } else {
        for (int i = rbeg; i < rbeg + rowsPerChunk; ++i)
            s += M[(size_t)i * ncols + j];
    }
    part[(size_t)chunk * ncols + j] = s;
}

// Stage 2: out[j] = sum_c part[c][j]  (fixed order -> deterministic)
__global__ void colsum_reduce_kernel(const float* __restrict__ part, float* __restrict__ out,
                                     int ncols, int nchunks) {
    const int j = blockIdx.x * blockDim.x + threadIdx.x;
    float s = 0.0f;
    for (int c = 0; c < nchunks; ++c) s += part[(size_t)c * ncols + j];
    out[j] = s;
}

// alpha = (w1 . Mh[:, -1]) / sum(Mh[:, -1]);  denom_p = sum(Mp[:, -1])
__global__ void __launch_bounds__(1024)
scalars_kernel(const float* __restrict__ Mh, const float* __restrict__ Mp,
               const float* __restrict__ w1, float* __restrict__ scal) {
    __shared__ float r0[1024], r1[1024], r2[1024];
    int t = threadIdx.x;
    float a = 0.f, b = 0.f, c = 0.f;
    for (int i = t; i < NH; i += 1024) {
        float mh = Mh[(size_t)i * HD + (HD - 1)];
        float mp = Mp[(size_t)i * PD + (PD - 1)];
        a += w1[i] * mh; b += mh; c += mp;
    }
    r0[t] = a; r1[t] = b; r2[t] = c; __syncthreads();
    for (int s = 512; s > 0; s >>= 1) {
        if (t < s) { r0[t] += r0[t + s]; r1[t] += r1[t + s]; r2[t] += r2[t + s]; }
        __syncthreads();
    }
    if (t == 0) { scal[0] = r0[0] / r1[0]; scal[1] = r2[0]; }
}

__global__ void sk0_kernel(const float* __restrict__ colsum, const float* __restrict__ scal,
                           float* __restrict__ sk) {
    int i = blockIdx.x * blockDim.x + threadIdx.x;
    if (i < HD) sk[i] = scal[0] * colsum[i];
}

// out = W @ x via V_WMMA_F32_16X16X4_F32 (full f32 precision).
// Each wave owns (16-row tile, K-half): B = x segment replicated across all 16
// N-columns so every output column equals the matvec result. Fragment layouts:
// A 16x4 f32: lanes 0-15 carry K=0,1 ; lanes 16-31 carry K=2,3 (2 VGPRs).
// C/D 16x16 f32: lane 0 holds rows 0..7 at N=0, lane 16 holds rows 8..15.
__global__ void __launch_bounds__(256)
wmma_matvec_kernel(const float* __restrict__ W, const float* __restrict__ x,
                   float* __restrict__ outA, float* __restrict__ outB, int kdim) {
    __shared__ float xs[2048];
    for (int i = threadIdx.x; i < kdim; i += 256) xs[i] = x[i];
    __syncthreads();

    const int lane  = threadIdx.x & 31;
    const int wave  = threadIdx.x >> 5;
    const int gw    = blockIdx.x * 8 + wave;
    const int tile  = gw >> 1;
    const int khalf = gw & 1;
    const int m     = lane & 15;
    const int koff  = (lane >> 4) * 2;          // 0 for lanes 0-15, 2 for lanes 16-31
    const int kh    = kdim >> 1;
    const int kbeg  = khalf * kh;
    const float* wrow = W + (size_t)(tile * 16 + m) * (size_t)kdim + kbeg;
    const float* xp   = xs + kbeg;

    v8f acc = {};
    #pragma unroll 8
    for (int k = 0; k < kh; k += 4) {
        v2f a = *(const v2f*)(wrow + k + koff);     // global_load_b64 (A fragment)
        v2f b = *(const v2f*)(xp + k + koff);       // ds_load_b64     (B fragment, N-replicated)
        acc = __builtin_amdgcn_wmma_f32_16x16x4_f32(
            /*neg_a=*/false, a, /*neg_b=*/false, b,
            /*c_mod=*/(short)0, acc, /*reuse_a=*/false, /*reuse_b=*/false);
    }
    float* out = khalf ? outB : outA;
    const int rowBase = tile * 16;
    if (lane == 0) {
        for (int j = 0; j < 8; ++j) out[rowBase + j] = acc[j];
    } else if (lane == 16) {
        for (int j = 0; j < 8; ++j) out[rowBase + 8 + j] = acc[j];
    }
}

// beta = (sk . W2T_mp) / denom_p
__global__ void __launch_bounds__(1024)
beta_kernel(const float* __restrict__ sk, const float* __restrict__ w2tmp,
            float* __restrict__ scal) {
    __shared__ float red[1024];
    int t = threadIdx.x;
    red[t] = sk[t] * w2tmp[t] + sk[t + 1024] * w2tmp[t + 1024];
    __syncthreads();
    for (int s = 512; s > 0; s >>= 1) {
        if (t < s) red[t] += red[t + s];
        __syncthreads();
    }
    if (t == 0) scal[2] = red[0] / scal[1];
}

// GRU gates + state update + feature projection + softmax accumulation.
__global__ void __launch_bounds__(1024)
gate_kernel(const float* __restrict__ wmA, const float* __restrict__ wmB,
            const float* __restrict__ ghA, const float* __restrict__ ghB,
            const float* __restrict__ mpsum, const float* __restrict__ W3,
            const float* __restrict__ scal, float* __restrict__ sk,
            float* __restrict__ accum, float* __restrict__ outFinal, int stepk) {
    __shared__ float r0[1024], r1[1024], r2[1024];
    const float beta = scal[2];
    const int t = threadIdx.x;
    float y0 = 0.f, y1 = 0.f, y2 = 0.f;
    #pragma unroll
    for (int rep = 0; rep < 2; ++rep) {
        const int h = t + rep * 1024;
        const float gir = beta * (wmA[h]          + wmB[h]);
        const float giz = beta * (wmA[HD + h]     + wmB[HD + h]);
        const float gin = beta * (wmA[2 * HD + h] + wmB[2 * HD + h]);
        const float ghr = ghA[h]          + ghB[h];
        const float ghz = ghA[HD + h]     + ghB[HD + h];
        const float ghn = ghA[2 * HD + h] + ghB[2 * HD + h];
        const float r  = 1.f / (1.f + expf(-(gir + ghr)));
        const float z  = 1.f / (1.f + expf(-(giz + ghz)));
        const float n  = tanhf(gin + r * ghn);
        const float sn = (1.f - z) * n + z * sk[h];
        sk[h] = sn;
        const float xk = beta * mpsum[h];
        const float ab = fabsf(sn - xk);
        const float pr = sn * xk;
        const float* w3a = W3 + (size_t)h * 3;
        const float* w3b = W3 + (size_t)(2048 + h) * 3;
        const float* w3c = W3 + (size_t)(4096 + h) * 3;
        const float* w3d = W3 + (size_t)(6144 + h) * 3;
        y0 += w3a[0] * sn + w3b[0] * xk + w3c[0] * ab + w3d[0] * pr;
        y1 += w3a[1] * sn + w3b[1] * xk + w3c[1] * ab + w3d[1] * pr;
        y2 += w3a[2] * sn + w3b[2] * xk + w3c[2] * ab + w3d[2] * pr;
    }
    r0[t] = y0; r1[t] = y1; r2[t] = y2; __syncthreads();
    for (int s = 512; s > 0; s >>= 1) {
        if (t < s) { r0[t] += r0[t + s]; r1[t] += r1[t + s]; r2[t] += r2[t + s]; }
        __syncthreads();
    }
    if (t == 0) {
        const float a = r0[0], b = r1[0], c = r2[0];
        const float mx = fmaxf(a, fmaxf(b, c));
        const float ea = expf(a - mx), eb = expf(b - mx), ec = expf(c - mx);
        const float inv = 1.f / (ea + eb + ec);
        const float a0 = (stepk == 0) ? 0.f : accum[0];
        const float a1 = (stepk == 0) ? 0.f : accum[1];
        const float a2 = (stepk == 0) ? 0.f : accum[2];
        const float n0 = a0 + ea * inv, n1 = a1 + eb * inv, n2 = a2 + ec * inv;
        accum[0] = n0; accum[1] = n1; accum[2] = n2;
        if (stepk == 7) {
            outFinal[0] = n0 * 0.125f;
            outFinal[1] = n1 * 0.125f;
            outFinal[2] = n2 * 0.125f;
        }
    }
}

extern "C" void kernel_launch(void* const* d_in, const int* in_sizes, int n_in,
                              void* d_out, int out_size, void* d_ws, size_t ws_size,
                              hipStream_t stream) {
    const float* M_h  = (const float*)d_in[0];
    const float* M_p  = (const float*)d_in[1];
    const float* w1   = (const float*)d_in[2];
    const float* W2   = (const float*)d_in[3];
    const float* W3   = (const float*)d_in[4];
    const float* W_ih = (const float*)d_in[5];
    const float* W_hh = (const float*)d_in[6];
    float* ws = (float*)d_ws;
    float* sk      = ws + WS_SK;
    float* colsumh = ws + WS_COLSUM_H;
    float* mpsum   = ws + WS_MP_SUM;
    float* w2tmp   = ws + WS_W2T_MP;
    float* wmA     = ws + WS_WMA;
    float* wmB     = ws + WS_WMB;
    float* ghA     = ws + WS_GHA;
    float* ghB     = ws + WS_GHB;
    float* accum   = ws + WS_ACCUM;
    float* scal    = ws + WS_SCAL;
    float* part    = ws + WS_PART;
    float* out     = (float*)d_out;

    const dim3 cgrid(HD / 256, NCHUNK);
    const int rpc = NH / NCHUNK;

    // Hoisted reductions (two-stage, deterministic, high-occupancy)
    wcolsum_part_kernel<<<cgrid, 256, 0, stream>>>(M_h, nullptr, 0, part, HD, rpc);
    colsum_reduce_kernel<<<HD / 256, 256, 0, stream>>>(part, colsumh, HD, NCHUNK);
    wcolsum_part_kernel<<<cgrid, 256, 0, stream>>>(M_p, nullptr, 0, part, PD, rpc);
    colsum_reduce_kernel<<<PD / 256, 256, 0, stream>>>(part, mpsum, PD, NCHUNK);
    wcolsum_part_kernel<<<cgrid, 256, 0, stream>>>(W2, M_p + (PD - 1), PD, part, HD, rpc);
    colsum_reduce_kernel<<<HD / 256, 256, 0, stream>>>(part, w2tmp, HD, NCHUNK);

    scalars_kernel<<<1, 1024, 0, stream>>>(M_h, M_p, w1, scal);
    sk0_kernel<<<HD / 256, 256, 0, stream>>>(colsumh, scal, sk);

    // WM = W_ih @ Mp_sum  (hoisted: gi = beta * WM every step); 2 K-halves per tile
    wmma_matvec_kernel<<<G3 / 64, 256, 0, stream>>>(W_ih, mpsum, wmA, wmB, PD);

    // K = 8 sequential GRU hops
    for (int k = 0; k < 8; ++k) {
        beta_kernel<<<1, 1024, 0, stream>>>(sk, w2tmp, scal);
        wmma_matvec_kernel<<<G3 / 64, 256, 0, stream>>>(W_hh, sk, ghA, ghB, HD);
        gate_kernel<<<1, 1024, 0, stream>>>(wmA, wmB, ghA, ghB, mpsum, W3, scal, sk,
                                            accum, out, k);
    }
}